// SegmentationAugmentation_59055800320542
// MI455X (gfx1250) — compile-verified
//
#include <hip/hip_runtime.h>
#include <hip/hip_bf16.h>
#include <stdint.h>

// SegmentationAugmentation for MI455X (gfx1250, wave32).
// Bandwidth-bound 3D affine trilinear resample: ~256 MB HBM traffic -> ~11us
// floor at 23.3 TB/s. Address math amortized over 8 batches x 2 volumes, and
// corner offsets kept as unsigned 32-bit byte offsets so loads lower to the
// GVS saddr form (uniform SGPR base + VGPR voffset) instead of per-lane
// 64-bit address arithmetic.

typedef __attribute__((ext_vector_type(2))) float v2f;
typedef __attribute__((ext_vector_type(8))) float v8f;

#define DIM    128
#define VOL    (DIM * DIM * DIM)      // 2,097,152 voxels per volume
#define NBATCH 8

__global__ __launch_bounds__(256)
void seg_aug_kernel(const float* __restrict__ inp,
                    const float* __restrict__ lab,
                    const float* __restrict__ xf,
                    float* __restrict__ out)
{
    __shared__ float s_theta[16];

    const unsigned tid = threadIdx.x;

    // ---- Stage the 4x4 transform into LDS via CDNA5 async global->LDS DMA.
    // One lane copies 64B with four b128 async transfers (ASYNCcnt-tracked),
    // then waits asynccnt==0 before the block barrier publishes the data.
    if (tid == 0) {
        unsigned lds_addr = (unsigned)(uintptr_t)(&s_theta[0]);
        unsigned long long gaddr = (unsigned long long)(uintptr_t)xf;
        asm volatile(
            "global_load_async_to_lds_b128 %0, %1, off\n\t"
            "global_load_async_to_lds_b128 %0, %1, off offset:16\n\t"
            "global_load_async_to_lds_b128 %0, %1, off offset:32\n\t"
            "global_load_async_to_lds_b128 %0, %1, off offset:48\n\t"
            "s_wait_asynccnt 0"
            :: "v"(lds_addr), "v"(gaddr) : "memory");
    }
    __syncthreads();

    // ---- Output voxel: wave spans 32 consecutive w -> coalesced stores, and
    // since theta's rotation is in x-y only, z' (the width sample coordinate)
    // is linear in w -> gathers along the wave are coalesced too.
    const unsigned vox = blockIdx.x * 256u + tid;   // grid is exact: VOL/256
    const int w = (int)(vox & 127u);
    const int h = (int)((vox >> 7) & 127u);
    const int d = (int)(vox >> 14);

    const float inv = 1.0f / 128.0f;
    const float px = (float)(2 * d + 1) * inv - 1.0f;  // x slot <- d
    const float py = (float)(2 * h + 1) * inv - 1.0f;  // y slot <- h
    const float pz = (float)(2 * w + 1) * inv - 1.0f;  // z slot <- w

    // ---- Affine transform via V_WMMA_F32_16X16X4_F32.
    // A (16x4): rows 0..2 = theta rows, rest zero. Per-lane: row m = lane&15,
    // K pair = (lane<16 ? K0,1 : K2,3). B (4x16): column n = point n; rows
    // 0,1 live in lanes 0-15, rows 2,3 in lanes 16-31. Result D rows 0..2
    // (VGPR 0..2, lanes 0-15) hold x',y',z' of the 16 points. Two WMMAs cover
    // the wave's 32 points; shuffles deliver group-1 coords to lanes 16-31.
    float cx, cy, cz;
#if __has_builtin(__builtin_amdgcn_wmma_f32_16x16x4_f32)
    {
        const int ln = (int)(tid & 31u);
        const int m  = ln & 15;
        const int k0 = (ln < 16) ? 0 : 2;
        v2f A;
        A[0] = (m < 3) ? s_theta[m * 4 + k0]     : 0.0f;
        A[1] = (m < 3) ? s_theta[m * 4 + k0 + 1] : 0.0f;

        // B0: points of lanes 0..15.  lanes<16 -> rows 0,1 (x,y of own point)
        //                             lanes>=16 -> rows 2,3 (z of point m, 1)
        const float zl = __shfl(pz, m);
        v2f B0;
        B0[0] = (ln < 16) ? px : zl;
        B0[1] = (ln < 16) ? py : 1.0f;

        // B1: points of lanes 16..31.
        const float xh = __shfl(px, 16 + m);
        const float yh = __shfl(py, 16 + m);
        v2f B1;
        B1[0] = (ln < 16) ? xh : pz;
        B1[1] = (ln < 16) ? yh : 1.0f;

        v8f Z = {};
        v8f D0 = __builtin_amdgcn_wmma_f32_16x16x4_f32(
            false, A, false, B0, (short)0, Z, false, false);
        v8f D1 = __builtin_amdgcn_wmma_f32_16x16x4_f32(
            false, A, false, B1, (short)0, Z, false, false);

        const float gx = __shfl(D1[0], m);
        const float gy = __shfl(D1[1], m);
        const float gz = __shfl(D1[2], m);
        cx = (ln < 16) ? D0[0] : gx;
        cy = (ln < 16) ? D0[1] : gy;
        cz = (ln < 16) ? D0[2] : gz;
    }
#else
    cx = fmaf(s_theta[0], px, fmaf(s_theta[1], py, fmaf(s_theta[2],  pz, s_theta[3])));
    cy = fmaf(s_theta[4], px, fmaf(s_theta[5], py, fmaf(s_theta[6],  pz, s_theta[7])));
    cz = fmaf(s_theta[8], px, fmaf(s_theta[9], py, fmaf(s_theta[10], pz, s_theta[11])));
#endif

    // ---- Un-normalize (align_corners=False) + border clamp; corner setup.
    const float ux = fminf(fmaxf((cx + 1.0f) * 64.0f - 0.5f, 0.0f), 127.0f); // depth
    const float uy = fminf(fmaxf((cy + 1.0f) * 64.0f - 0.5f, 0.0f), 127.0f); // height
    const float uz = fminf(fmaxf((cz + 1.0f) * 64.0f - 0.5f, 0.0f), 127.0f); // width

    const float fdx = floorf(ux), fdy = floorf(uy), fdz = floorf(uz);
    const float fx = ux - fdx, fy = uy - fdy, fz = uz - fdz;
    const unsigned di = (unsigned)(int)fdx;
    const unsigned hi = (unsigned)(int)fdy;
    const unsigned wi = (unsigned)(int)fdz;
    const unsigned di1 = (di + 1u < 127u) ? di + 1u : 127u;
    const unsigned hi1 = (hi + 1u < 127u) ? hi + 1u : 127u;
    const unsigned wi1 = (wi + 1u < 127u) ? wi + 1u : 127u;

    // Unsigned BYTE offsets (max ~8MB, fits 32-bit) -> saddr+voffset loads.
    const unsigned r00 = ((di  << 7) + hi ) << 9;   // (di*128+hi)*128*4
    const unsigned r01 = ((di  << 7) + hi1) << 9;
    const unsigned r10 = ((di1 << 7) + hi ) << 9;
    const unsigned r11 = ((di1 << 7) + hi1) << 9;
    const unsigned cw0 = wi  << 2;
    const unsigned cw1 = wi1 << 2;

    const unsigned o000 = r00 + cw0, o001 = r00 + cw1;
    const unsigned o010 = r01 + cw0, o011 = r01 + cw1;
    const unsigned o100 = r10 + cw0, o101 = r10 + cw1;
    const unsigned o110 = r11 + cw0, o111 = r11 + cw1;

    const float W000 = (1.0f - fx) * (1.0f - fy) * (1.0f - fz);
    const float W001 = (1.0f - fx) * (1.0f - fy) * fz;
    const float W010 = (1.0f - fx) * fy * (1.0f - fz);
    const float W011 = (1.0f - fx) * fy * fz;
    const float W100 = fx * (1.0f - fy) * (1.0f - fz);
    const float W101 = fx * (1.0f - fy) * fz;
    const float W110 = fx * fy * (1.0f - fz);
    const float W111 = fx * fy * fz;

    const unsigned obyte = vox << 2;   // output byte offset within a volume

    // ---- Gather loop: address math reused 16x (8 batches x 2 volumes).
    // Sources are L2-resident (128MB < 192MB); outputs are streamed with
    // non-temporal stores so they don't evict the gather working set.
    #pragma unroll
    for (int b = 0; b < NBATCH; ++b) {
        const char* __restrict__ ib = (const char*)inp + (size_t)b * (VOL * 4u);
        const char* __restrict__ lb = (const char*)lab + (size_t)b * (VOL * 4u);

        #define LD(base, off) (*(const float*)((base) + (off)))
        float s = W000 * LD(ib, o000) + W001 * LD(ib, o001)
                + W010 * LD(ib, o010) + W011 * LD(ib, o011)
                + W100 * LD(ib, o100) + W101 * LD(ib, o101)
                + W110 * LD(ib, o110) + W111 * LD(ib, o111);

        float t = W000 * LD(lb, o000) + W001 * LD(lb, o001)
                + W010 * LD(lb, o010) + W011 * LD(lb, o011)
                + W100 * LD(lb, o100) + W101 * LD(lb, o101)
                + W110 * LD(lb, o110) + W111 * LD(lb, o111);
        #undef LD

        float* po = (float*)((char*)out + (size_t)b * (VOL * 4u) + obyte);
        float* pl = (float*)((char*)out + (size_t)NBATCH * (VOL * 4u)
                                        + (size_t)b * (VOL * 4u) + obyte);
        __builtin_nontemporal_store(s, po);
        __builtin_nontemporal_store((t > 0.5f) ? 1.0f : 0.0f, pl);
    }
}

extern "C" void kernel_launch(void* const* d_in, const int* in_sizes, int n_in,
                              void* d_out, int out_size, void* d_ws, size_t ws_size,
                              hipStream_t stream) {
    const float* input_g   = (const float*)d_in[0];   // [8,1,128,128,128] f32
    const float* label_g   = (const float*)d_in[1];   // [8,1,128,128,128] f32
    const float* transform = (const float*)d_in[2];   // [4,4] f32
    float* out = (float*)d_out;                        // aug_inp (16.7M) ++ mask (16.7M)

    (void)in_sizes; (void)n_in; (void)out_size; (void)d_ws; (void)ws_size;

    seg_aug_kernel<<<VOL / 256, 256, 0, stream>>>(input_g, label_g, transform, out);
}